// HeteroGNN_55559696941685
// MI455X (gfx1250) — compile-verified
//
#include <hip/hip_runtime.h>
#include <hip/hip_bf16.h>
#include <math.h>

typedef __attribute__((ext_vector_type(2))) float v2f;
typedef __attribute__((ext_vector_type(8))) float v8f;

#define N_NODES 50000
#define N_EDGES 800000
#define DD_IN   64
#define DD_HID  128
#define DD_OUT  300

// ---------------------------------------------------------------- utilities
__global__ void zero_f32(float* __restrict__ p, long n) {
  long i = (long)blockIdx.x * blockDim.x + threadIdx.x;
  if (i < n) p[i] = 0.0f;
}

__global__ void degree_kernel(const int* __restrict__ dst, float* __restrict__ deg, int E) {
  int e = blockIdx.x * blockDim.x + threadIdx.x;
  if (e < E) atomicAdd(&deg[dst[e]], 1.0f);
}

// one thread per (edge, feature); consecutive lanes -> consecutive addresses
template <int LOG2D>
__global__ void scatter_kernel(const int* __restrict__ src, const int* __restrict__ dst,
                               const float* __restrict__ feat, float* __restrict__ sums,
                               long total) {
  long i = (long)blockIdx.x * blockDim.x + threadIdx.x;
  if (i >= total) return;
  const int e = (int)(i >> LOG2D);
  const int f = (int)(i & ((1 << LOG2D) - 1));
  const float v = feat[((long)src[e] << LOG2D) + f];
  atomicAdd(&sums[((long)dst[e] << LOG2D) + f], v);
}

// ------------------------------------------------- fused SAGE GEMM via WMMA
// out[16 rows x NOUT] = relu?( (sums/deg) @ Wl + bias + xin @ Wr )
// FP32 end-to-end via V_WMMA_F32_16X16X4_F32. NOUT is compile-time so all
// weight accesses are immediate offsets from one lane-constant base; loads
// are prefetched in chunks of 8 k-steps so 16 WMMAs run per load burst.
template <int K, int NOUT, bool RELU>
__global__ void __launch_bounds__(256)
sage_gemm_wmma(const float* __restrict__ sums, const float* __restrict__ deg,
               const float* __restrict__ xin,
               const float* __restrict__ Wl, const float* __restrict__ Wr,
               const float* __restrict__ bias,
               float* __restrict__ out) {
  __shared__ float As[16][K + 4];  // aggregated (already /deg) tile
  __shared__ float Xs[16][K + 4];  // root-feature tile
  const int rowBase = blockIdx.x * 16;  // N_NODES % 16 == 0, no row guard

  // float4 staging (row stride K+4 floats = 16B-aligned; pad keeps A-frag
  // reads conflict-free: (K+4) mod 64 = 4 -> 16 distinct banks per half-wave)
  constexpr int VPR = K / 4;  // float4 per row
  for (int idx = threadIdx.x; idx < 16 * VPR; idx += 256) {
    const int r = idx / VPR, kv = idx - r * VPR;
    const int row = rowBase + r;
    const float inv = 1.0f / fmaxf(deg[row], 1.0f);
    float4 s = ((const float4*)(sums + (long)row * K))[kv];
    float4 xv = ((const float4*)(xin + (long)row * K))[kv];
    s.x *= inv; s.y *= inv; s.z *= inv; s.w *= inv;
    *(float4*)&As[r][kv * 4] = s;
    *(float4*)&Xs[r][kv * 4] = xv;
  }
  __syncthreads();

  const int lane  = threadIdx.x & 31;
  const int wave  = threadIdx.x >> 5;
  const int m     = lane & 15;           // M (A-frag) / N (B-frag) index
  const int khalf = (lane >> 4) << 1;    // lanes 16-31 hold K+2,K+3

  // lane-constant LDS bases: asp[k] -> immediate DS offsets
  const float* __restrict__ asp = &As[m][khalf];
  const float* __restrict__ xsp = &Xs[m][khalf];

  constexpr int NCT   = (NOUT + 15) / 16;  // col tiles
  constexpr int NSTEP = K / 4;             // wmma pairs over K
  constexpr int CH    = 8;                 // k-steps per prefetch chunk
  static_assert(NSTEP % CH == 0, "chunking");

  for (int c = wave; c < NCT; c += 8) {   // branch uniform per wave
    const int n  = c * 16 + m;
    const int nc = (n < NOUT) ? n : (NOUT - 1);   // clamp, no guard:
    // dead columns corrupt only their own (never-stored) lanes
    const float bv = bias[nc];
    v8f acc = {bv, bv, bv, bv, bv, bv, bv, bv};

    // lane-constant global bases: wl2[k*NOUT] -> immediate VMEM offsets
    const float* __restrict__ wl2 = Wl + nc + khalf * NOUT;
    const float* __restrict__ wr2 = Wr + nc + khalf * NOUT;

#pragma unroll
    for (int k0 = 0; k0 < NSTEP; k0 += CH) {
      v2f al[CH], ar[CH], bl[CH], br[CH];
      // ---- load burst: 32 global + 32 DS loads, all immediate offsets ----
#pragma unroll
      for (int j = 0; j < CH; ++j) {
        const int k = (k0 + j) * 4;
        bl[j].x = wl2[k * NOUT];
        bl[j].y = wl2[(k + 1) * NOUT];
        br[j].x = wr2[k * NOUT];
        br[j].y = wr2[(k + 1) * NOUT];
        al[j].x = asp[k];
        al[j].y = asp[k + 1];
        ar[j].x = xsp[k];
        ar[j].y = xsp[k + 1];
      }
      // ---- 16 back-to-back WMMAs ----
#pragma unroll
      for (int j = 0; j < CH; ++j) {
        acc = __builtin_amdgcn_wmma_f32_16x16x4_f32(false, al[j], false, bl[j],
                                                    (short)0, acc, false, false);
        acc = __builtin_amdgcn_wmma_f32_16x16x4_f32(false, ar[j], false, br[j],
                                                    (short)0, acc, false, false);
      }
    }

    if (RELU) {
#pragma unroll
      for (int r = 0; r < 8; ++r) acc[r] = fmaxf(acc[r], 0.0f);
    }
    if (n < NOUT) {
      const int rb = rowBase + ((lane >> 4) << 3);  // C/D layout: M=r / M=r+8
      float* __restrict__ op = out + (long)rb * NOUT + n;
#pragma unroll
      for (int r = 0; r < 8; ++r) op[r * NOUT] = acc[r];
    }
  }
}

// ------------------------------------------------- dropout (JAX threefry2x32)
__device__ __forceinline__ unsigned rotl32(unsigned x, int r) {
  return (x << r) | (x >> (32 - r));
}
__device__ __forceinline__ void tf_rounds4(unsigned& x0, unsigned& x1,
                                           int r0, int r1, int r2, int r3) {
  x0 += x1; x1 = rotl32(x1, r0); x1 ^= x0;
  x0 += x1; x1 = rotl32(x1, r1); x1 ^= x0;
  x0 += x1; x1 = rotl32(x1, r2); x1 ^= x0;
  x0 += x1; x1 = rotl32(x1, r3); x1 ^= x0;
}

__global__ void dropout_threefry(float* __restrict__ h, long total) {
  const long half = total >> 1;
  const long tid  = (long)blockIdx.x * blockDim.x + threadIdx.x;
  if (tid >= half) return;
  const unsigned k0 = 0u, k1 = 42u;              // jax.random.key(42)
  const unsigned k2 = k0 ^ k1 ^ 0x1BD11BDAu;
  unsigned x0 = (unsigned)tid + k0;              // counts split in halves
  unsigned x1 = (unsigned)(tid + half) + k1;
  tf_rounds4(x0, x1, 13, 15, 26, 6);  x0 += k1; x1 += k2 + 1u;
  tf_rounds4(x0, x1, 17, 29, 16, 24); x0 += k2; x1 += k0 + 2u;
  tf_rounds4(x0, x1, 13, 15, 26, 6);  x0 += k0; x1 += k1 + 3u;
  tf_rounds4(x0, x1, 17, 29, 16, 24); x0 += k1; x1 += k2 + 4u;
  tf_rounds4(x0, x1, 13, 15, 26, 6);  x0 += k2; x1 += k0 + 5u;
  const float u0 = __uint_as_float((x0 >> 9) | 0x3f800000u) - 1.0f;
  const float u1 = __uint_as_float((x1 >> 9) | 0x3f800000u) - 1.0f;
  const float a = h[tid];
  const float b = h[tid + half];
  h[tid]        = (u0 < 0.5f) ? a * 2.0f : 0.0f;
  h[tid + half] = (u1 < 0.5f) ? b * 2.0f : 0.0f;
}

// ------------------------------------------------- row-wise log_softmax (wave32)
__global__ void logsoftmax_rows(float* __restrict__ out, int M, int Ncols) {
  const int row  = blockIdx.x * (blockDim.x >> 5) + (threadIdx.x >> 5);
  const int lane = threadIdx.x & 31;
  if (row >= M) return;
  float* p = out + (long)row * Ncols;
  float mx = -INFINITY;
  for (int j = lane; j < Ncols; j += 32) mx = fmaxf(mx, p[j]);
  for (int o = 16; o; o >>= 1) mx = fmaxf(mx, __shfl_xor(mx, o, 32));
  float s = 0.0f;
  for (int j = lane; j < Ncols; j += 32) s += expf(p[j] - mx);
  for (int o = 16; o; o >>= 1) s += __shfl_xor(s, o, 32);
  const float lse = mx + logf(s);
  for (int j = lane; j < Ncols; j += 32) p[j] = p[j] - lse;
}

// ---------------------------------------------------------------- launcher
extern "C" void kernel_launch(void* const* d_in, const int* in_sizes, int n_in,
                              void* d_out, int out_size, void* d_ws, size_t ws_size,
                              hipStream_t stream) {
  const float* x   = (const float*)d_in[0];
  const int*   ei  = (const int*)d_in[1];
  const float* Wl1 = (const float*)d_in[2];
  const float* Wr1 = (const float*)d_in[3];
  const float* b1  = (const float*)d_in[4];
  const float* Wl2 = (const float*)d_in[5];
  const float* Wr2 = (const float*)d_in[6];
  const float* b2  = (const float*)d_in[7];
  float* out = (float*)d_out;

  const int* src = ei;
  const int* dst = ei + N_EDGES;

  // workspace layout: deg | agg1 | h | agg2  (~64.2 MB)
  float* deg  = (float*)d_ws;
  float* agg1 = deg  + N_NODES;
  float* h    = agg1 + (size_t)N_NODES * DD_IN;
  float* agg2 = h    + (size_t)N_NODES * DD_HID;

  // --- layer 1 aggregation ---
  {
    long n = N_NODES + (long)N_NODES * DD_IN;   // deg + agg1 contiguous
    zero_f32<<<(unsigned)((n + 255) / 256), 256, 0, stream>>>(deg, n);
  }
  degree_kernel<<<(N_EDGES + 255) / 256, 256, 0, stream>>>(dst, deg, N_EDGES);
  {
    long total = (long)N_EDGES * DD_IN;
    scatter_kernel<6><<<(unsigned)((total + 255) / 256), 256, 0, stream>>>(
        src, dst, x, agg1, total);
  }
  // --- layer 1 GEMM + bias + relu ---
  sage_gemm_wmma<DD_IN, DD_HID, true><<<N_NODES / 16, 256, 0, stream>>>(
      agg1, deg, x, Wl1, Wr1, b1, h);
  // --- dropout ---
  {
    long total = (long)N_NODES * DD_HID;
    long half  = total >> 1;
    dropout_threefry<<<(unsigned)((half + 255) / 256), 256, 0, stream>>>(h, total);
  }
  // --- layer 2 aggregation ---
  {
    long n = (long)N_NODES * DD_HID;
    zero_f32<<<(unsigned)((n + 255) / 256), 256, 0, stream>>>(agg2, n);
  }
  {
    long total = (long)N_EDGES * DD_HID;
    scatter_kernel<7><<<(unsigned)((total + 255) / 256), 256, 0, stream>>>(
        src, dst, h, agg2, total);
  }
  // --- layer 2 GEMM + bias (no relu), write to d_out ---
  sage_gemm_wmma<DD_HID, DD_OUT, false><<<N_NODES / 16, 256, 0, stream>>>(
      agg2, deg, h, Wl2, Wr2, b2, out);
  // --- log_softmax in place ---
  logsoftmax_rows<<<N_NODES / 8, 256, 0, stream>>>(out, N_NODES, DD_OUT);

  (void)in_sizes; (void)n_in; (void)out_size; (void)ws_size;
}